// UmbrellaSurfaceConstructor_v2_33285996544417
// MI455X (gfx1250) — compile-verified
//
#include <hip/hip_runtime.h>
#include <hip/hip_bf16.h>
#include <math.h>

#define KNN 9
#define TBATCH 4  // candidate tiles (of 16 points) processed per barrier pair

typedef float v2f __attribute__((ext_vector_type(2)));
typedef float v8f __attribute__((ext_vector_type(8)));

#if __has_builtin(__builtin_amdgcn_wmma_f32_16x16x4_f32)
#define HAVE_WMMA_F32X4 1
#else
#define HAVE_WMMA_F32X4 0
#endif

// ---------------------------------------------------------------------------
// Pack points as float4 {x, y, z, |p|^2} so the squared-norm rides in the
// 4th WMMA K-slot.
// ---------------------------------------------------------------------------
__global__ void pack_pts(const float* __restrict__ c, float4* __restrict__ p4, int N) {
  int i = blockIdx.x * blockDim.x + threadIdx.x;
  if (i < N) {
    float x = c[3 * i], y = c[3 * i + 1], z = c[3 * i + 2];
    p4[i] = make_float4(x, y, z, x * x + y * y + z * z);
  }
}

// Branch-free (after unroll) insertion into a sorted ascending top-K list kept
// entirely in VGPRs. Strict '<' keeps earlier (smaller) indices on ties.
__device__ __forceinline__ void topk_insert(float (&ld)[KNN], int (&li)[KNN],
                                            float d, int idx) {
  if (d >= ld[KNN - 1]) return;
#pragma unroll
  for (int p = KNN - 1; p >= 0; --p) {
    float pd = (p > 0) ? ld[p - 1] : -3.4e38f;
    int   pi = (p > 0) ? li[p - 1] : 0;
    bool lt  = d < ld[p];
    bool ltp = d < pd;
    float nv = lt ? (ltp ? pd : d) : ld[p];
    int   ni = lt ? (ltp ? pi : idx) : li[p];
    ld[p] = nv;
    li[p] = ni;
  }
}

// ---------------------------------------------------------------------------
// kNN: one wave32 per 16-query tile. Distances via V_WMMA_F32_16X16X4_F32:
//   A row m = [-2*qx, -2*qy, -2*qz, 1]   (16x4, f32)
//   B col n = [ px,    py,    pz, |p|^2] (4x16, f32)
//   D[m][n] = |p|^2 - 2 q.p  ==  |q - p|^2 - |q|^2   (row-constant shift,
//   ordering-preserving -> identical top-K indices as true distances).
// TBATCH tiles are computed back-to-back (4 WMMAs) per barrier pair; the
// per-row scan reads the staged distances with ds_load_b128.
// ---------------------------------------------------------------------------
__global__ __launch_bounds__(32) void knn_wmma(const float4* __restrict__ p4,
                                               int* __restrict__ gi, int N) {
  __shared__ float4 tile4[TBATCH * 64];   // TBATCH x 16 x 16 floats
  __shared__ float  md[16 * KNN];
  __shared__ int    mi[16 * KNN];
#if !HAVE_WMMA_F32X4
  __shared__ float4 qs[16];
#endif
  float* tile = (float*)tile4;

  const int lane = (int)threadIdx.x;
  const int r    = lane & 15;
  const int half = lane >> 4;
  const int qbase = blockIdx.x * 16;

  float4 q = p4[qbase + r];
#if HAVE_WMMA_F32X4
  // A-matrix 16x4 f32 layout: lanes 0-15 -> {K0,K1}, lanes 16-31 -> {K2,K3}
  v2f a;
  a.x = half ? (-2.0f * q.z) : (-2.0f * q.x);
  a.y = half ? 1.0f          : (-2.0f * q.y);
#else
  if (half == 0) qs[r] = q;
  __syncthreads();
#endif

  float ld[KNN];
  int   li[KNN];
#pragma unroll
  for (int k = 0; k < KNN; ++k) { ld[k] = 3.4e38f; li[k] = 0x7fffffff; }

  const int ntiles = N >> 4;
  const int cbase  = half * 8;

  for (int t0 = 0; t0 < ntiles; t0 += TBATCH) {
    if (t0 + TBATCH < ntiles)
      __builtin_prefetch((const void*)(p4 + (t0 + TBATCH) * 16 + r), 0, 3);

    // ---- compute TBATCH distance tiles with back-to-back WMMAs ----
#pragma unroll
    for (int u = 0; u < TBATCH; ++u) {
      float4 p = p4[(t0 + u) * 16 + r];
#if HAVE_WMMA_F32X4
      // B-matrix 4x16 f32 layout: lanes 0-15 -> {K0,K1}, lanes 16-31 -> {K2,K3}
      v2f b;
      b.x = half ? p.z : p.x;
      b.y = half ? p.w : p.y;
      v8f acc = {0.f, 0.f, 0.f, 0.f, 0.f, 0.f, 0.f, 0.f};
      acc = __builtin_amdgcn_wmma_f32_16x16x4_f32(
          /*neg_a=*/false, a, /*neg_b=*/false, b,
          /*c_mod=*/(short)0, acc, /*reuse_a=*/false, /*reuse_b=*/false);
      // C/D layout: lanes 0-15 hold rows 0..7 (vgpr i = row i), 16-31 rows 8..15
#pragma unroll
      for (int i = 0; i < 8; ++i)
        tile[u * 256 + (i + half * 8) * 16 + r] = acc[i];
#else
#pragma unroll
      for (int i = 0; i < 8; ++i) {
        float4 qq = qs[i + half * 8];
        float d = p.w - 2.0f * (qq.x * p.x + qq.y * p.y + qq.z * p.z);
        tile[u * 256 + (i + half * 8) * 16 + r] = d;
      }
#endif
    }
    __syncthreads();

    // ---- scan: lane L owns cols 0..7, lane L+16 cols 8..15 of row r ----
#pragma unroll
    for (int u = 0; u < TBATCH; ++u) {
      const int v4 = u * 64 + r * 4 + half * 2;  // float4 index of row segment
      float4 d0 = tile4[v4];
      float4 d1 = tile4[v4 + 1];
      const int ibase = (t0 + u) * 16 + cbase;
      topk_insert(ld, li, d0.x, ibase + 0);
      topk_insert(ld, li, d0.y, ibase + 1);
      topk_insert(ld, li, d0.z, ibase + 2);
      topk_insert(ld, li, d0.w, ibase + 3);
      topk_insert(ld, li, d1.x, ibase + 4);
      topk_insert(ld, li, d1.y, ibase + 5);
      topk_insert(ld, li, d1.z, ibase + 6);
      topk_insert(ld, li, d1.w, ibase + 7);
    }
    __syncthreads();
  }

  // Merge the two half-row lists (high-half list folded into low-half list).
  if (half) {
#pragma unroll
    for (int k = 0; k < KNN; ++k) { md[r * KNN + k] = ld[k]; mi[r * KNN + k] = li[k]; }
  }
  __syncthreads();
  if (!half) {
#pragma unroll
    for (int k = 0; k < KNN; ++k) topk_insert(ld, li, md[r * KNN + k], mi[r * KNN + k]);
#pragma unroll
    for (int k = 0; k < KNN; ++k) gi[(qbase + r) * KNN + k] = li[k];
  }
}

// ---------------------------------------------------------------------------
// Umbrella normals: one thread per point, all-register K=9 pipeline.
// ---------------------------------------------------------------------------
__global__ void umbrella_normals(const float* __restrict__ c,
                                 const int* __restrict__ gi,
                                 float* __restrict__ out, int N) {
  int i = blockIdx.x * blockDim.x + threadIdx.x;
  if (i >= N) return;

  const float cx = c[3 * i], cy = c[3 * i + 1], cz = c[3 * i + 2];

  float gx[KNN], gy[KNN], gz[KNN], ph[KNN];
  int ok[KNN];
#pragma unroll
  for (int k = 0; k < KNN; ++k) {
    int j = gi[i * KNN + k];
    float x = c[3 * j]     - cx;
    float y = c[3 * j + 1] - cy;
    float z = c[3 * j + 2] - cz;
    gx[k] = x; gy[k] = y; gz[k] = z;
    // rot = g @ ROT ; phi = atan2(rot.y, rot.x)
    float rx =  0.5f * x + 0.7071f * y - 0.5f * z;
    float ry = -0.5f * x + 0.7071f * y + 0.5f * z;
    ph[k] = atan2f(ry, rx);
    ok[k] = k;
  }

  // Stable sort by (phi, original index) with an optimal 25-CE network;
  // vectors ride along so no dynamic register indexing is needed.
#define CSWAP(A, B)                                                            \
  {                                                                            \
    bool sw = (ph[A] > ph[B]) || (ph[A] == ph[B] && ok[A] > ok[B]);            \
    float tp = ph[A], tx = gx[A], ty = gy[A], tz = gz[A]; int tk = ok[A];      \
    ph[A] = sw ? ph[B] : ph[A]; gx[A] = sw ? gx[B] : gx[A];                    \
    gy[A] = sw ? gy[B] : gy[A]; gz[A] = sw ? gz[B] : gz[A];                    \
    ok[A] = sw ? ok[B] : ok[A];                                                \
    ph[B] = sw ? tp : ph[B]; gx[B] = sw ? tx : gx[B];                          \
    gy[B] = sw ? ty : gy[B]; gz[B] = sw ? tz : gz[B];                          \
    ok[B] = sw ? tk : ok[B];                                                   \
  }
  CSWAP(0, 3) CSWAP(1, 7) CSWAP(2, 5) CSWAP(4, 8)
  CSWAP(0, 7) CSWAP(2, 4) CSWAP(3, 8) CSWAP(5, 6)
  CSWAP(0, 2) CSWAP(1, 3) CSWAP(4, 5) CSWAP(7, 8)
  CSWAP(1, 4) CSWAP(3, 6) CSWAP(5, 7)
  CSWAP(0, 1) CSWAP(2, 4) CSWAP(3, 5) CSWAP(6, 8)
  CSWAP(2, 3) CSWAP(4, 5) CSWAP(6, 7)
  CSWAP(1, 2) CSWAP(3, 4) CSWAP(5, 6)
#undef CSWAP

  float ux[KNN], uy[KNN], uz[KNN], nn[KNN];
  bool bad[KNN];
#pragma unroll
  for (int k = 0; k < KNN; ++k) {
    int k2 = (k + 1) % KNN;  // constant after unroll
    float nx = gy[k] * gz[k2] - gz[k] * gy[k2];
    float ny = gz[k] * gx[k2] - gx[k] * gz[k2];
    float nz = gx[k] * gy[k2] - gy[k] * gx[k2];
    float n  = sqrtf(nx * nx + ny * ny + nz * nz);
    nn[k]  = n;
    bad[k] = n < 1e-12f;
    float dv = bad[k] ? 1.0f : n;
    ux[k] = nx / dv; uy[k] = ny / dv; uz[k] = nz / dv;
  }

  // Keep first triangle's x-component positive.
  float pos = (ux[0] > 0.0f) ? 1.0f : -1.0f;
#pragma unroll
  for (int k = 0; k < KNN; ++k) { ux[k] *= pos; uy[k] *= pos; uz[k] *= pos; }

  // check_nan_umb: replace degenerate normals by first valid one (index 0 if none).
  float fx = ux[0], fy = uy[0], fz = uz[0];
  bool found = !bad[0];
#pragma unroll
  for (int k = 1; k < KNN; ++k) {
    bool take = !found && !bad[k];
    fx = take ? ux[k] : fx; fy = take ? uy[k] : fy; fz = take ? uz[k] : fz;
    found = found || !bad[k];
  }
#pragma unroll
  for (int k = 0; k < KNN; ++k) {
    ux[k] = bad[k] ? fx : ux[k];
    uy[k] = bad[k] ? fy : uy[k];
    uz[k] = bad[k] ? fz : uz[k];
  }

  // softmax(area / 1e-4) weighted aggregation, max-subtracted for stability.
  float mx = -3.4e38f;
#pragma unroll
  for (int k = 0; k < KNN; ++k) mx = fmaxf(mx, 0.5f * nn[k]);
  float wsum = 0.0f, px = 0.0f, py = 0.0f, pz = 0.0f;
#pragma unroll
  for (int k = 0; k < KNN; ++k) {
    float w = expf((0.5f * nn[k] - mx) * 1e4f);
    wsum += w;
    px += ux[k] * w; py += uy[k] * w; pz += uz[k] * w;
  }
  float inv = 1.0f / wsum;
  out[3 * i]     = px * inv;
  out[3 * i + 1] = py * inv;
  out[3 * i + 2] = pz * inv;
}

// ---------------------------------------------------------------------------
extern "C" void kernel_launch(void* const* d_in, const int* in_sizes, int n_in,
                              void* d_out, int out_size, void* d_ws, size_t ws_size,
                              hipStream_t stream) {
  const float* center = (const float*)d_in[0];
  const int N = in_sizes[0] / 3;  // [N, 3] float32

  float4* p4 = (float4*)d_ws;                       // N * 16 bytes of scratch
  int*    gi = (int*)d_out;                         // group_idx: N*9 int32
  float*  pn = (float*)d_out + (size_t)N * KNN;     // point_normal: N*3 f32

  pack_pts<<<(N + 255) / 256, 256, 0, stream>>>(center, p4, N);
  knn_wmma<<<N / 16, 32, 0, stream>>>(p4, gi, N);
  umbrella_normals<<<(N + 255) / 256, 256, 0, stream>>>(center, gi, pn, N);
}